// MultiBandGraphBuilder_68264210203160
// MI455X (gfx1250) — compile-verified
//
#include <hip/hip_runtime.h>
#include <math.h>

// ---------------------------------------------------------------------------
// MultiBandGraphBuilder for MI455X (gfx1250, wave32, WMMA).
// FP32 WMMA (16x16x4) used for the two GEMM-heavy stages: the 116x116
// correlation covariances (512 of them) and the 116->64->116 intra MLP.
// Everything else is bandwidth-bound scatter/reduce work.
// ---------------------------------------------------------------------------

typedef float v2f __attribute__((ext_vector_type(2)));
typedef float v8f __attribute__((ext_vector_type(8)));

__device__ __forceinline__ float sigmoidf(float x) { return 1.0f / (1.0f + expf(-x)); }

__device__ __forceinline__ v8f wmma4(v2f a, v2f b, v8f c) {
  // D = A(16x4,f32) x B(4x16,f32) + C(16x16,f32)
  return __builtin_amdgcn_wmma_f32_16x16x4_f32(false, a, false, b, (short)0, c, false, false);
}

constexpr int Bn = 128, Nn = 116, Kn = 4, Fn = 72, Sn = 8;
constexpr int G  = Bn * Kn;        // 512 intra groups
constexpr int KN = Kn * Nn;        // 464
constexpr int NP = 128;            // N padded to 8x16 tiles
constexpr int LDC = 73;            // LDS row stride for centered data (odd -> no bank conflict)

constexpr long long U_SZ     = (long long)Bn * KN * KN;       // 27,557,888
constexpr long long NODE_SZ  = (long long)Bn * KN * Fn;       //  4,276,224
constexpr long long GF_SZ    = (long long)Bn * 128;           //     16,384
constexpr long long INTRA_SZ = (long long)G * Nn * Nn;        //  6,889,472
constexpr long long INTER_SZ = (long long)Bn * Nn * Kn * Kn;  //    237,568

constexpr long long OFF_NODE  = U_SZ;
constexpr long long OFF_GRAPH = OFF_NODE + NODE_SZ;
constexpr long long OFF_INTRA = OFF_GRAPH + GF_SZ;
constexpr long long OFF_INTER = OFF_INTRA + INTRA_SZ;
constexpr long long OFF_SP    = OFF_INTER + INTER_SZ;

// workspace layout (floats)
constexpr long long WS_CORR = 0;                               // corr, later reused as raw intra MLP out
constexpr long long WS_TAU  = WS_CORR + INTRA_SZ;              // 10th-largest |corr| per row
constexpr long long WS_ACC  = WS_TAU + (long long)G * Nn;      // accumulators
constexpr long long WS_GF   = WS_ACC + 1024;                   // per-b graph feature staging (128*150)

// accumulator slots
constexpr int A_DEG = 0, A_DEGSQ = 128, A_CLUS = 256, A_CLUSSQ = 384;
constexpr int A_DENS = 512, A_STREN = 640, A_INZ = 768, A_JNZ = 769;

// ---------------------------------------------------------------------------
__global__ void zero_unified_kernel(float4* __restrict__ u) {
  long long idx = (long long)blockIdx.x * blockDim.x + threadIdx.x;
  if (idx < U_SZ / 4) u[idx] = float4{0.f, 0.f, 0.f, 0.f};
}

__global__ void zero_acc_kernel(float* __restrict__ acc) {
  if (threadIdx.x < 1024) acc[threadIdx.x] = 0.f;
}

// node_features[b, k*N+n, f] = band_features[b, n, k, f]
__global__ void node_features_kernel(const float* __restrict__ bf, float* __restrict__ out) {
  long long idx = (long long)blockIdx.x * blockDim.x + threadIdx.x;
  if (idx >= NODE_SZ) return;
  int f = (int)(idx % Fn);
  long long r = idx / Fn;
  int nk = (int)(r % KN);
  int b  = (int)(r / KN);
  int k = nk / Nn, n = nk % Nn;
  out[OFF_NODE + idx] = bf[(((long long)b * Nn + n) * Kn + k) * Fn + f];
}

// ---------------------------------------------------------------------------
// 512 blocks, 8 waves each: corr[g] = clip(cov / (std_i*std_j + 1e-8), -1, 1)
// cov computed with V_WMMA_F32_16X16X4_F32 over 8x8 16x16 tiles, 18 K-steps.
__global__ __launch_bounds__(256) void corr_intra_kernel(const float* __restrict__ bf,
                                                         float* __restrict__ corr) {
  __shared__ float c[NP * LDC];
  __shared__ float stdv[NP];
  const int g = blockIdx.x;            // g = b*K + k
  const int b = g >> 2, k = g & 3;
  const int t = threadIdx.x;

  // load x[i][f] = bf[b, i, k, f]; zero-pad rows 116..127
  for (int idx = t; idx < NP * Fn; idx += 256) {
    int i = idx / Fn, f = idx % Fn;
    float v = 0.f;
    if (i < Nn) v = bf[(((long long)b * Nn + i) * Kn + k) * Fn + f];
    c[i * LDC + f] = v;
  }
  __syncthreads();

  if (t < NP) {
    float s = 0.f;
    for (int f = 0; f < Fn; ++f) s += c[t * LDC + f];
    float mu = s * (1.0f / Fn);
    float var = 0.f;
    for (int f = 0; f < Fn; ++f) {
      float v = c[t * LDC + f] - mu;
      c[t * LDC + f] = v;
      var += v * v;
    }
    stdv[t] = sqrtf(fmaxf(var, 1e-8f));
  }
  __syncthreads();

  const int wave = t >> 5, lane = t & 31, half = lane >> 4, lr = lane & 15;
  const int mt = wave;  // 8 waves = 8 tile rows
  for (int nt = 0; nt < 8; ++nt) {
    v8f acc = {0, 0, 0, 0, 0, 0, 0, 0};
    for (int kk = 0; kk < Fn; kk += 4) {
      const int ka = kk + half * 2;  // A/B VGPR0 holds K=0|2, VGPR1 K=1|3 split on lane half
      v2f a, bb;
      const int arow = mt * 16 + lr;
      a.x = c[arow * LDC + ka];
      a.y = c[arow * LDC + ka + 1];
      const int brow = nt * 16 + lr;  // B[k][n] = c[n][k]
      bb.x = c[brow * LDC + ka];
      bb.y = c[brow * LDC + ka + 1];
      acc = wmma4(a, bb, acc);
    }
#pragma unroll
    for (int r = 0; r < 8; ++r) {
      const int M  = mt * 16 + r + half * 8;
      const int Nc = nt * 16 + lr;
      if (M < Nn && Nc < Nn) {
        float cr = acc[r] / (stdv[M] * stdv[Nc] + 1e-8f);
        cr = fminf(fmaxf(cr, -1.0f), 1.0f);
        corr[(long long)g * (Nn * Nn) + M * Nn + Nc] = cr;
      }
    }
  }
}

// ---------------------------------------------------------------------------
// One wave per row: tau = 10th largest |corr| (matches top_k selection).
__global__ __launch_bounds__(256) void tau_kernel(const float* __restrict__ corr,
                                                  float* __restrict__ tau) {
  const int row  = blockIdx.x * 8 + (threadIdx.x >> 5);  // 59392 rows exactly
  const int lane = threadIdx.x & 31;
  const float* rp = corr + (long long)row * Nn;
  float v[4];
#pragma unroll
  for (int s = 0; s < 4; ++s) {
    int j = lane * 4 + s;
    v[s] = (j < Nn) ? fabsf(rp[j]) : -1.f;
  }
  float kth = 0.f;
  for (int it = 0; it < 10; ++it) {
    float lmax = fmaxf(fmaxf(v[0], v[1]), fmaxf(v[2], v[3]));
    float wmax = lmax;
    for (int off = 16; off; off >>= 1) wmax = fmaxf(wmax, __shfl_xor(wmax, off, 32));
    kth = wmax;
    int cand = (lmax == wmax) ? lane : 64;
    int mn = cand;
    for (int off = 16; off; off >>= 1) mn = min(mn, __shfl_xor(mn, off, 32));
    if (lane == mn) {
      for (int s = 0; s < 4; ++s) {
        if (v[s] == wmax) { v[s] = -2.f; break; }
      }
    }
  }
  if (lane == 0) tau[row] = kth;
}

// sc = relu(sym(corr*tmask*kmask)) = relu(0.5*c_ij*t_ij*(k_ij + k_ji)); staged in out[intra]
__global__ void sc_kernel(const float* __restrict__ corr, const float* __restrict__ tau,
                          const float* __restrict__ thadj, float* __restrict__ out) {
  long long idx = (long long)blockIdx.x * blockDim.x + threadIdx.x;
  if (idx >= INTRA_SZ) return;
  const int g = (int)(idx / (Nn * Nn));
  const int rem = (int)(idx % (Nn * Nn));
  const int i = rem / Nn, j = rem % Nn;
  const float th = 0.1f + sigmoidf(thadj[0]) * 0.2f;
  const float cv = corr[idx];
  const float a = fabsf(cv);
  const float tm = (a > th) ? 1.f : 0.f;
  const float ki = (a >= tau[g * Nn + i]) ? 1.f : 0.f;
  const float kj = (a >= tau[g * Nn + j]) ? 1.f : 0.f;
  out[OFF_INTRA + idx] = fmaxf(cv * tm * 0.5f * (ki + kj), 0.f);
}

// ---------------------------------------------------------------------------
// Fused intra MLP, 32 rows per block: H = relu(SC@W1^T + b1); OUT = sigmoid(H@W2^T + b2)
// Both GEMMs in fp32 WMMA. W1/W2 time-share one LDS buffer. Raw OUT -> ws (pre-sym).
__global__ __launch_bounds__(256) void mlp_intra_kernel(const float* __restrict__ sc_in,
                                                        const float* __restrict__ W1,
                                                        const float* __restrict__ b1,
                                                        const float* __restrict__ W2,
                                                        const float* __restrict__ b2,
                                                        float* __restrict__ raw_out) {
  __shared__ float scs[32 * 117];
  __shared__ float wsh[128 * 65];  // phase1: w1[h*117+i] (<=7488); phase2: w2[j*65+h] (8320)
  __shared__ float hs[32 * 65];
  __shared__ float b1s[64];
  __shared__ float b2s[116];
  const int t = threadIdx.x;
  const long long row0 = (long long)blockIdx.x * 32;

  for (int idx = t; idx < 32 * Nn; idx += 256) {
    int r = idx / Nn, i = idx % Nn;
    scs[r * 117 + i] = sc_in[(row0 + r) * Nn + i];
  }
  for (int idx = t; idx < 64 * Nn; idx += 256) {
    int h = idx / Nn, i = idx % Nn;
    wsh[h * 117 + i] = W1[idx];
  }
  if (t < 64) b1s[t] = b1[t];
  if (t < Nn) b2s[t] = b2[t];
  __syncthreads();

  const int wave = t >> 5, lane = t & 31, half = lane >> 4, lr = lane & 15;

  {  // GEMM1: 2x4 tiles of 16x16, K=116 (29 wmma steps); one tile per wave
    const int mt = wave >> 2, nt = wave & 3;
    v8f acc = {0, 0, 0, 0, 0, 0, 0, 0};
    for (int kk = 0; kk < Nn; kk += 4) {
      const int ka = kk + half * 2;
      v2f a, bb;
      a.x = scs[(mt * 16 + lr) * 117 + ka];
      a.y = scs[(mt * 16 + lr) * 117 + ka + 1];
      bb.x = wsh[(nt * 16 + lr) * 117 + ka];  // B[k=i][n=h] = W1[h][i]
      bb.y = wsh[(nt * 16 + lr) * 117 + ka + 1];
      acc = wmma4(a, bb, acc);
    }
#pragma unroll
    for (int r = 0; r < 8; ++r) {
      const int M = mt * 16 + r + half * 8;
      const int h = nt * 16 + lr;
      hs[M * 65 + h] = fmaxf(acc[r] + b1s[h], 0.f);
    }
  }
  __syncthreads();

  for (int idx = t; idx < 128 * 64; idx += 256) {  // load W2 (zero-pad rows 116..127)
    int j = idx / 64, h = idx % 64;
    wsh[j * 65 + h] = (j < Nn) ? W2[j * 64 + h] : 0.f;
  }
  __syncthreads();

  for (int tt = wave; tt < 16; tt += 8) {  // GEMM2: 2x8 tiles, K=64 (16 wmma steps)
    const int mt = tt >> 3, nt = tt & 7;
    v8f acc = {0, 0, 0, 0, 0, 0, 0, 0};
    for (int kk = 0; kk < 64; kk += 4) {
      const int ka = kk + half * 2;
      v2f a, bb;
      a.x = hs[(mt * 16 + lr) * 65 + ka];
      a.y = hs[(mt * 16 + lr) * 65 + ka + 1];
      bb.x = wsh[(nt * 16 + lr) * 65 + ka];  // B[k=h][n=j] = W2[j][h]
      bb.y = wsh[(nt * 16 + lr) * 65 + ka + 1];
      acc = wmma4(a, bb, acc);
    }
#pragma unroll
    for (int r = 0; r < 8; ++r) {
      const int M = mt * 16 + r + half * 8;
      const int j = nt * 16 + lr;
      if (j < Nn) raw_out[(row0 + M) * Nn + j] = sigmoidf(acc[r] + b2s[j]);
    }
  }
}

// intra_adj = sym(raw); also scatter into unified diagonal blocks
__global__ void sym_scatter_intra_kernel(const float* __restrict__ raw, float* __restrict__ out) {
  long long idx = (long long)blockIdx.x * blockDim.x + threadIdx.x;
  if (idx >= INTRA_SZ) return;
  const int g = (int)(idx / (Nn * Nn));
  const int rem = (int)(idx % (Nn * Nn));
  const int i = rem / Nn, j = rem % Nn;
  const float a = 0.5f * (raw[idx] + raw[(long long)g * (Nn * Nn) + j * Nn + i]);
  out[OFF_INTRA + idx] = a;
  const int b = g >> 2, k = g & 3;
  out[(long long)b * KN * KN + (long long)(k * Nn + i) * KN + (k * Nn + j)] = a;
}

// per-row deg/tri -> per-b accumulators (wave per row). A is exactly symmetric post-sym.
__global__ __launch_bounds__(256) void intra_stats_kernel(const float* __restrict__ out,
                                                          float* __restrict__ acc) {
  const int row  = blockIdx.x * 8 + (threadIdx.x >> 5);
  const int lane = threadIdx.x & 31;
  const float* A = out + OFF_INTRA + (long long)row * Nn;
  float deg = 0.f, tri = 0.f;
  for (int j = lane; j < Nn; j += 32) {
    float v = A[j];
    deg += (v > 0.f) ? 1.f : 0.f;
    tri += v * v;  // A_ij * A_ji (bitwise-symmetric)
  }
  for (int off = 16; off; off >>= 1) {
    deg += __shfl_xor(deg, off, 32);
    tri += __shfl_xor(tri, off, 32);
  }
  if (lane == 0) {
    const int b = (row / Nn) >> 2;
    const float clus = tri / (deg * deg + 1e-8f);
    atomicAdd(&acc[A_DEG + b], deg);
    atomicAdd(&acc[A_DEGSQ + b], deg * deg);
    atomicAdd(&acc[A_CLUS + b], clus);
    atomicAdd(&acc[A_CLUSSQ + b], clus * clus);
    atomicAdd(&acc[A_INZ], deg);
  }
}

// ---------------------------------------------------------------------------
// Inter path: one thread per (b,n). 4x4 corr(y, F=72) + corr(attn^T, S=8),
// fuse, threshold, relu, 4->32->4 MLP, sym; writes inter_adj + unified off-diag.
__global__ __launch_bounds__(256) void inter_kernel(const float* __restrict__ bf,
                                                    const float* __restrict__ aw,
                                                    const float* __restrict__ thadj,
                                                    const float* __restrict__ fusew,
                                                    const float* __restrict__ W1,
                                                    const float* __restrict__ b1,
                                                    const float* __restrict__ W2,
                                                    const float* __restrict__ b2,
                                                    float* __restrict__ out,
                                                    float* __restrict__ acc) {
  const int id = blockIdx.x * blockDim.x + threadIdx.x;
  if (id >= Bn * Nn) return;
  const int b = id / Nn, n = id % Nn;

  float base[4][4], acorr[4][4];
  {  // corr of y (K x F)
    const float* yb = bf + ((long long)b * Nn + n) * Kn * Fn;
    float s[4] = {0, 0, 0, 0}, p[10];
    for (int q = 0; q < 10; ++q) p[q] = 0.f;
    for (int f = 0; f < Fn; ++f) {
      float yv[4];
#pragma unroll
      for (int kx = 0; kx < 4; ++kx) yv[kx] = yb[kx * Fn + f];
      int pi = 0;
#pragma unroll
      for (int kx = 0; kx < 4; ++kx) {
        s[kx] += yv[kx];
#pragma unroll
        for (int jx = 0; jx <= kx; ++jx) p[pi++] += yv[jx] * yv[kx];
      }
    }
    float stds[4];
#pragma unroll
    for (int kx = 0; kx < 4; ++kx)
      stds[kx] = sqrtf(fmaxf(p[kx * (kx + 1) / 2 + kx] - s[kx] * s[kx] * (1.f / Fn), 1e-8f));
#pragma unroll
    for (int kx = 0; kx < 4; ++kx)
#pragma unroll
      for (int jx = 0; jx <= kx; ++jx) {
        float cov = p[kx * (kx + 1) / 2 + jx] - s[jx] * s[kx] * (1.f / Fn);
        float c = fminf(fmaxf(cov / (stds[jx] * stds[kx] + 1e-8f), -1.f), 1.f);
        base[jx][kx] = base[kx][jx] = c;
      }
  }
  {  // corr of attn^T (K x S)
    const float* ab = aw + ((long long)b * Nn + n) * Sn * Kn;
    float s[4] = {0, 0, 0, 0}, p[10];
    for (int q = 0; q < 10; ++q) p[q] = 0.f;
    for (int ss = 0; ss < Sn; ++ss) {
      float av[4];
#pragma unroll
      for (int kx = 0; kx < 4; ++kx) av[kx] = ab[ss * Kn + kx];
      int pi = 0;
#pragma unroll
      for (int kx = 0; kx < 4; ++kx) {
        s[kx] += av[kx];
#pragma unroll
        for (int jx = 0; jx <= kx; ++jx) p[pi++] += av[jx] * av[kx];
      }
    }
    float stds[4];
#pragma unroll
    for (int kx = 0; kx < 4; ++kx)
      stds[kx] = sqrtf(fmaxf(p[kx * (kx + 1) / 2 + kx] - s[kx] * s[kx] * (1.f / Sn), 1e-8f));
#pragma unroll
    for (int kx = 0; kx < 4; ++kx)
#pragma unroll
      for (int jx = 0; jx <= kx; ++jx) {
        float cov = p[kx * (kx + 1) / 2 + jx] - s[jx] * s[kx] * (1.f / Sn);
        float c = fminf(fmaxf(cov / (stds[jx] * stds[kx] + 1e-8f), -1.f), 1.f);
        acorr[jx][kx] = acorr[kx][jx] = c;
      }
  }

  const float fw = sigmoidf(fusew[0]);
  const float th2 = 0.2f + sigmoidf(thadj[0]) * 0.2f;
  float sc2[4][4];
#pragma unroll
  for (int i = 0; i < 4; ++i)
#pragma unroll
    for (int j = 0; j < 4; ++j) {
      float m = fw * base[i][j] + (1.f - fw) * acorr[i][j];  // symmetric -> sym() is identity
      m = (fabsf(m) > th2) ? m : 0.f;
      sc2[i][j] = fmaxf(m, 0.f);
    }

  float interm[4][4];
#pragma unroll
  for (int r = 0; r < 4; ++r) {
    float h[32];
#pragma unroll
    for (int o = 0; o < 32; ++o) {
      float z = b1[o];
#pragma unroll
      for (int i = 0; i < 4; ++i) z += W1[o * 4 + i] * sc2[r][i];
      h[o] = fmaxf(z, 0.f);
    }
#pragma unroll
    for (int j = 0; j < 4; ++j) {
      float z = b2[j];
#pragma unroll
      for (int o = 0; o < 32; ++o) z += W2[j * 32 + o] * h[o];
      interm[r][j] = sigmoidf(z);
    }
  }

  float ia[4][4];
  float cnt = 0.f, sum = 0.f;
#pragma unroll
  for (int i = 0; i < 4; ++i)
#pragma unroll
    for (int j = 0; j < 4; ++j) {
      float v = 0.5f * (interm[i][j] + interm[j][i]);
      ia[i][j] = v;
      cnt += (v > 0.f) ? 1.f : 0.f;
      sum += v;
      out[OFF_INTER + (long long)id * 16 + i * 4 + j] = v;
    }
  atomicAdd(&acc[A_DENS + b], cnt * (1.f / 16.f));
  atomicAdd(&acc[A_STREN + b], sum * (1.f / 16.f));
  atomicAdd(&acc[A_JNZ], cnt);

#pragma unroll
  for (int k1 = 0; k1 < 4; ++k1)
#pragma unroll
    for (int k2 = 0; k2 < 4; ++k2)
      if (k1 != k2)
        out[(long long)b * KN * KN + (long long)(k1 * Nn + n) * KN + (k2 * Nn + n)] = ia[k1][k2];
}

// per-(b,f) mean/std over the 464 (n,k) samples -> gf[0..143]
__global__ __launch_bounds__(256) void node_stats_kernel(const float* __restrict__ bf,
                                                         float* __restrict__ gf) {
  const int id   = blockIdx.x * 8 + (threadIdx.x >> 5);  // 9216 = 1152*8
  const int lane = threadIdx.x & 31;
  const int b = id / Fn, f = id % Fn;
  float s = 0.f, ss = 0.f;
  for (int t = lane; t < Nn * Kn; t += 32) {
    int n = t >> 2, k = t & 3;
    float v = bf[(((long long)b * Nn + n) * Kn + k) * Fn + f];
    s += v;
    ss += v * v;
  }
  for (int off = 16; off; off >>= 1) {
    s += __shfl_xor(s, off, 32);
    ss += __shfl_xor(ss, off, 32);
  }
  if (lane == 0) {
    gf[b * 150 + f] = s * (1.f / 464.f);
    gf[b * 150 + 72 + f] = sqrtf(fmaxf((ss - s * s * (1.f / 464.f)) * (1.f / 463.f), 0.f));
  }
}

// one block per b: finish gf[144..149], then graph_features = relu(gf @ comp_W^T + b)
__global__ __launch_bounds__(160) void graph_feature_kernel(const float* __restrict__ gf_in,
                                                            const float* __restrict__ acc,
                                                            const float* __restrict__ cW,
                                                            const float* __restrict__ cb,
                                                            float* __restrict__ out) {
  __shared__ float gfs[152];
  const int b = blockIdx.x, t = threadIdx.x;
  if (t < 144) gfs[t] = gf_in[b * 150 + t];
  else if (t == 144) gfs[144] = acc[A_DEG + b] * (1.f / 464.f);
  else if (t == 145) {
    float s = acc[A_DEG + b], ss = acc[A_DEGSQ + b];
    gfs[145] = sqrtf(fmaxf((ss - s * s * (1.f / 464.f)) * (1.f / 463.f), 0.f));
  } else if (t == 146) gfs[146] = acc[A_CLUS + b] * (1.f / 464.f);
  else if (t == 147) {
    float s = acc[A_CLUS + b], ss = acc[A_CLUSSQ + b];
    gfs[147] = sqrtf(fmaxf((ss - s * s * (1.f / 464.f)) * (1.f / 463.f), 0.f));
  } else if (t == 148) gfs[148] = acc[A_DENS + b] * (1.f / 116.f);
  else if (t == 149) gfs[149] = acc[A_STREN + b] * (1.f / 116.f);
  __syncthreads();
  if (t < 128) {
    float z = cb[t];
    for (int i = 0; i < 150; ++i) z += cW[t * 150 + i] * gfs[i];
    out[OFF_GRAPH + b * 128 + t] = fmaxf(z, 0.f);
  }
}

__global__ void sparsity_kernel(const float* __restrict__ acc, float* __restrict__ out) {
  const float it = (float)INTRA_SZ, jt = (float)INTER_SZ;
  const float isp = 1.f - acc[A_INZ] / it;
  const float jsp = 1.f - acc[A_JNZ] / jt;
  out[OFF_SP + 0] = isp;
  out[OFF_SP + 1] = jsp;
  out[OFF_SP + 2] = (isp * it + jsp * jt) / (it + jt);
}

// ---------------------------------------------------------------------------
extern "C" void kernel_launch(void* const* d_in, const int* in_sizes, int n_in,
                              void* d_out, int out_size, void* d_ws, size_t ws_size,
                              hipStream_t stream) {
  (void)in_sizes; (void)n_in; (void)out_size; (void)ws_size;
  const float* bf   = (const float*)d_in[0];   // band_features (B,N,K,F)
  const float* aw   = (const float*)d_in[1];   // attn_weights  (B,N,S,K)
  const float* thI  = (const float*)d_in[2];   // intra_thresh_adj
  const float* thJ  = (const float*)d_in[3];   // inter_thresh_adj
  const float* fuse = (const float*)d_in[4];   // attn_fuse_weight
  const float* iW1  = (const float*)d_in[5];   // (64,116)
  const float* ib1  = (const float*)d_in[6];
  const float* iW2  = (const float*)d_in[7];   // (116,64)
  const float* ib2  = (const float*)d_in[8];
  const float* jW1  = (const float*)d_in[9];   // (32,4)
  const float* jb1  = (const float*)d_in[10];
  const float* jW2  = (const float*)d_in[11];  // (4,32)
  const float* jb2  = (const float*)d_in[12];
  const float* cW   = (const float*)d_in[13];  // (128,150)
  const float* cb   = (const float*)d_in[14];

  float* out  = (float*)d_out;
  float* ws   = (float*)d_ws;
  float* corr = ws + WS_CORR;
  float* tau  = ws + WS_TAU;
  float* acc  = ws + WS_ACC;
  float* gf   = ws + WS_GF;

  zero_unified_kernel<<<(int)((U_SZ / 4 + 255) / 256), 256, 0, stream>>>((float4*)out);
  zero_acc_kernel<<<1, 1024, 0, stream>>>(acc);
  node_features_kernel<<<(int)((NODE_SZ + 255) / 256), 256, 0, stream>>>(bf, out);
  corr_intra_kernel<<<G, 256, 0, stream>>>(bf, corr);
  tau_kernel<<<G * Nn / 8, 256, 0, stream>>>(corr, tau);
  sc_kernel<<<(int)((INTRA_SZ + 255) / 256), 256, 0, stream>>>(corr, tau, thI, out);
  mlp_intra_kernel<<<G * Nn / 32, 256, 0, stream>>>(out + OFF_INTRA, iW1, ib1, iW2, ib2, corr);
  sym_scatter_intra_kernel<<<(int)((INTRA_SZ + 255) / 256), 256, 0, stream>>>(corr, out);
  intra_stats_kernel<<<G * Nn / 8, 256, 0, stream>>>(out, acc);
  inter_kernel<<<(Bn * Nn + 255) / 256, 256, 0, stream>>>(bf, aw, thJ, fuse, jW1, jb1, jW2, jb2, out, acc);
  node_stats_kernel<<<Bn * Fn / 8, 256, 0, stream>>>(bf, gf);
  graph_feature_kernel<<<Bn, 160, 0, stream>>>(gf, acc, cW, cb, out);
  sparsity_kernel<<<1, 1, 0, stream>>>(acc, out);
}